// WeightedMessagePassing_14474039787719
// MI455X (gfx1250) — compile-verified
//
#include <hip/hip_runtime.h>

// WeightedMessagePassing for MI455X (gfx1250, wave32).
//   out = concat(h, segment_sum(w * h[src], dst)) @ W^T + b
// Phase 1: zero agg. Phase 2: edge scatter, native f32 global atomics (L2-resident).
// Phase 3: fp32 WMMA GEMM [N,128]x[128,64] via V_WMMA_F32_16X16X4_F32, with
//          gfx1250 async global->LDS staging (GLOBAL_LOAD_ASYNC_TO_LDS_B128).

typedef float v2f __attribute__((ext_vector_type(2)));
typedef float v8f __attribute__((ext_vector_type(8)));
typedef int   wmp_v4i __attribute__((vector_size(16)));   // b128 payload type

#define DDIM   64
#define K2     128          // 2*D (concat width)
#define LDSTR  132          // padded LDS row stride (floats): 528B, 16B-aligned,
                            // bank-shift 4/row -> conflict-free b64 frag loads

#if __has_builtin(__builtin_amdgcn_global_load_async_to_lds_b128) && \
    __has_builtin(__builtin_amdgcn_s_wait_asynccnt)
#define WMP_ASYNC_LDS 1
#endif

#define WMP_AS1 __attribute__((address_space(1)))
#define WMP_AS3 __attribute__((address_space(3)))

// Copy 16B global -> LDS. Async (ASYNCcnt-tracked, no VGPR round-trip) when
// the gfx1250 builtins are available; plain b128 load/store otherwise.
__device__ __forceinline__ void wmp_copy16(const float* gsrc, float* ldst) {
#ifdef WMP_ASYNC_LDS
  __builtin_amdgcn_global_load_async_to_lds_b128(
      (WMP_AS1 wmp_v4i*)(uintptr_t)gsrc,
      (WMP_AS3 wmp_v4i*)(uint32_t)(uintptr_t)ldst, /*offset=*/0, /*cpol=*/0);
#else
  *(float4*)ldst = *(const float4*)gsrc;
#endif
}

__device__ __forceinline__ void wmp_stage_fence() {
#ifdef WMP_ASYNC_LDS
  __builtin_amdgcn_s_wait_asynccnt(0);
#endif
  __syncthreads();
}

__global__ void __launch_bounds__(256)
wmp_zero_kernel(float* __restrict__ p, int n4) {
  int i = blockIdx.x * blockDim.x + threadIdx.x;
  if (i < n4) ((float4*)p)[i] = make_float4(0.f, 0.f, 0.f, 0.f);
}

// One wave32 per edge; lane l handles features {2l, 2l+1}.
__global__ void __launch_bounds__(256)
wmp_scatter_kernel(const float* __restrict__ h, const float* __restrict__ w,
                   const int* __restrict__ src, const int* __restrict__ dst,
                   float* __restrict__ agg, int E) {
  int e    = (int)((blockIdx.x * blockDim.x + threadIdx.x) >> 5);
  int lane = threadIdx.x & 31;
  if (e >= E) return;
  int   s  = src[e];
  int   d  = dst[e];
  float we = w[e];
  const float2 hv = *(const float2*)(h + (size_t)s * DDIM + lane * 2);
  float* ap = agg + (size_t)d * DDIM + lane * 2;
  unsafeAtomicAdd(ap,     we * hv.x);   // global_atomic_add_f32
  unsafeAtomicAdd(ap + 1, we * hv.y);
}

// 8 waves/block; wave w owns output rows [row0 + 16w, row0 + 16w + 16).
// A-frag (16x4 f32): lanes 0-15 hold K-pair {k,k+1}, lanes 16-31 hold {k+2,k+3}.
// B-frag (4x16 f32): mirrored layout; B[k][n] = W^T[k][n] = W[n][k], so storing W
// row-major in LDS gives contiguous float2 fragment loads for both A and B.
__global__ void __launch_bounds__(256)
wmp_gemm_kernel(const float* __restrict__ h, const float* __restrict__ agg,
                const float* __restrict__ W, const float* __restrict__ b,
                float* __restrict__ out, int n) {
  __shared__ float sA[128 * LDSTR];   // concat tile: 128 rows x 128 cols
  __shared__ float sW[64 * LDSTR];    // W ([64][128] row-major == W^T col data)

  const int  tid      = threadIdx.x;
  const int  row0     = blockIdx.x * 128;
  const bool fullTile = (row0 + 128 <= n);

  // Stage W (coalesced 16B chunks).
  for (int c = tid; c < 64 * 32; c += 256) {
    int r = c >> 5, c4 = (c & 31) * 4;
    wmp_copy16(&W[r * K2 + c4], &sW[r * LDSTR + c4]);
  }
  // Stage concat(h, agg) tile; zero-pad rows past N (last block only).
  if (fullTile) {
    for (int c = tid; c < 128 * 32; c += 256) {
      int r = c >> 5, c4 = (c & 31) * 4;
      const float* gsrc = (c4 < 64) ? &h[(size_t)(row0 + r) * DDIM + c4]
                                    : &agg[(size_t)(row0 + r) * DDIM + (c4 - 64)];
      wmp_copy16(gsrc, &sA[r * LDSTR + c4]);
    }
  } else {
    for (int c = tid; c < 128 * 32; c += 256) {
      int r = c >> 5, c4 = (c & 31) * 4;
      int g = row0 + r;
      if (g < n) {
        const float* gsrc = (c4 < 64) ? &h[(size_t)g * DDIM + c4]
                                      : &agg[(size_t)g * DDIM + (c4 - 64)];
        wmp_copy16(gsrc, &sA[r * LDSTR + c4]);
      } else {
        *(float4*)&sA[r * LDSTR + c4] = make_float4(0.f, 0.f, 0.f, 0.f);
      }
    }
  }
  wmp_stage_fence();   // s_wait_asynccnt 0 (if async) + barrier

  const int wid  = tid >> 5;
  const int lane = tid & 31;
  const int idx  = lane & 15;   // M for A-frag, N for B-frag / C lane column
  const int half = lane >> 4;   // K-pair selector

  // Init accumulators with bias: lane's column N = nt*16 + idx for all 8 rows.
  v8f acc[4];
#pragma unroll
  for (int nt = 0; nt < 4; ++nt) {
    float bn = b[nt * 16 + idx];
    v8f vi = {bn, bn, bn, bn, bn, bn, bn, bn};
    acc[nt] = vi;
  }

  const float* aRow = &sA[(wid * 16 + idx) * LDSTR + 2 * half];
#pragma unroll 4
  for (int k = 0; k < K2; k += 4) {
    v2f av = *(const v2f*)(aRow + k);
#pragma unroll
    for (int nt = 0; nt < 4; ++nt) {
      v2f bv = *(const v2f*)(&sW[(nt * 16 + idx) * LDSTR + 2 * half + k]);
      acc[nt] = __builtin_amdgcn_wmma_f32_16x16x4_f32(
          /*neg_a=*/false, av, /*neg_b=*/false, bv,
          /*c_mod=*/(short)0, acc[nt], /*reuse_a=*/false, /*reuse_b=*/false);
    }
  }

  // C/D layout: VGPR r, lanes 0-15 -> M=r, lanes 16-31 -> M=8+r; N = lane&15.
  float* obase = out + (size_t)(row0 + wid * 16 + half * 8) * DDIM + idx;
  if (fullTile) {
#pragma unroll
    for (int nt = 0; nt < 4; ++nt)
#pragma unroll
      for (int r = 0; r < 8; ++r)
        obase[(size_t)r * DDIM + nt * 16] = acc[nt][r];
  } else {
#pragma unroll
    for (int nt = 0; nt < 4; ++nt)
#pragma unroll
      for (int r = 0; r < 8; ++r) {
        int grow = row0 + wid * 16 + half * 8 + r;
        if (grow < n) out[(size_t)grow * DDIM + nt * 16 + idx] = acc[nt][r];
      }
  }
}

extern "C" void kernel_launch(void* const* d_in, const int* in_sizes, int n_in,
                              void* d_out, int out_size, void* d_ws, size_t ws_size,
                              hipStream_t stream) {
  const float* h   = (const float*)d_in[0];
  const float* w   = (const float*)d_in[1];
  const int*   src = (const int*)d_in[2];
  const int*   dst = (const int*)d_in[3];
  const float* W   = (const float*)d_in[4];
  const float* b   = (const float*)d_in[5];
  float*       out = (float*)d_out;

  const int N = in_sizes[0] / DDIM;   // 100000
  const int E = in_sizes[2];          // 1600000

  // agg scratch: prefer d_ws; fall back to in-place d_out (safe: gemm stages
  // its agg tile into LDS before overwriting those same rows).
  float* agg = (ws_size >= (size_t)N * DDIM * sizeof(float)) ? (float*)d_ws : out;

  int n4 = (N * DDIM) / 4;
  wmp_zero_kernel<<<(n4 + 255) / 256, 256, 0, stream>>>(agg, n4);

  int scatterBlocks = (E + 7) / 8;    // 8 edges (waves) per 256-thread block
  wmp_scatter_kernel<<<scatterBlocks, 256, 0, stream>>>(h, w, src, dst, agg, E);

  int gemmBlocks = (N + 127) / 128;
  wmp_gemm_kernel<<<gemmBlocks, 256, 0, stream>>>(h, agg, W, b, out, N);
}